// STMGG_20633022890742
// MI455X (gfx1250) — compile-verified
//
#include <hip/hip_runtime.h>
#include <hip/hip_bf16.h>

typedef __attribute__((ext_vector_type(2))) float v2f;
typedef __attribute__((ext_vector_type(8))) float v8f;

#define B_     48
#define S_     64
#define N_     64
#define HOR_   12
#define SN_    4096          // S_*N_
#define LDA    65            // padded LDS stride (bank-conflict-free row reads)
#define KPART  24            // split-K partitions in contract kernel (3 g x 8)

// ---------------------------------------------------------------------------
// One 16x16 output tile of a 64x64x64 fp32 GEMM via V_WMMA_F32_16X16X4_F32.
// A-frag (16x4 f32): lanes 0-15 -> K=0,1 ; lanes 16-31 -> K=2,3 (M = lane&15).
// B-frag (4x16 f32): same K split; N = lane&15.
// C/D (16x16 f32, 8 VGPRs): VGPR r -> M = r (+8 for lanes 16-31), N = lane&15.
// ---------------------------------------------------------------------------
__device__ __forceinline__ void wmma_tile_f32(
    const float* __restrict__ A, int lda,
    const float* __restrict__ B, int ldb, bool transB,
    float* __restrict__ D, int ldd,
    int m0, int n0, int lane, float scale, const float* __restrict__ colBias)
{
    v8f acc = {};
    const int half = (lane < 16) ? 0 : 2;
    const int l15  = lane & 15;
    const int m = m0 + l15;
    const int n = n0 + l15;
    for (int k0 = 0; k0 < 64; k0 += 4) {
        v2f a, b;
        a.x = A[m * lda + k0 + half];
        a.y = A[m * lda + k0 + half + 1];
        if (!transB) {
            b.x = B[(k0 + half) * ldb + n];
            b.y = B[(k0 + half + 1) * ldb + n];
        } else {                     // B[k][n] = Bsrc[n][k]
            b.x = B[n * ldb + k0 + half];
            b.y = B[n * ldb + k0 + half + 1];
        }
        acc = __builtin_amdgcn_wmma_f32_16x16x4_f32(
                  false, a, false, b, (short)0, acc, false, false);
    }
    float bias = colBias ? colBias[n] : 0.0f;
#pragma unroll
    for (int r = 0; r < 8; ++r) {
        int mm = m0 + r + ((lane < 16) ? 0 : 8);
        D[mm * ldd + n] = acc[r] * scale + bias;
    }
}

// ---------------------------------------------------------------------------
// Kernel B: exact pairwise DTW over a[n][t] = x4[0,t,n,0], plus avg_adj.
// Block i handles row i; thread j handles pair (i,j). Rolling DP row lives in
// per-thread LDS column rowS[s*64 + j] (bank == j -> conflict-free).
// ---------------------------------------------------------------------------
__global__ __launch_bounds__(64) void dtw_kernel(
    const float* __restrict__ x,
    float* __restrict__ dtw_adj, float* __restrict__ avg_adj)
{
    __shared__ float aS[64 * 65];
    __shared__ float rowS[64 * 64];
    __shared__ float spS[64];
    const int i = blockIdx.x;
    const int j = threadIdx.x;

    for (int idx = j; idx < 4096; idx += 64) {
        int n = idx & 63, t = idx >> 6;
        aS[n * 65 + t] = x[(t * 64 + n) * 64];   // x4[0,t,n,0]
    }
    __syncthreads();

    float s = 0.f;
    for (int t = 0; t < 64; ++t) s += aS[j * 65 + t];
    spS[j] = s * (1.f / 64.f);
    __syncthreads();

    avg_adj[i * 64 + j] = (i == j) ? 0.f : 0.5f / (spS[i] + spS[j]);

    // row t = 0: cumulative sum of |a_i[0] - a_j[s]|
    const float ai0 = aS[i * 65];
    float acc = 0.f;
    for (int ss = 0; ss < 64; ++ss) {
        acc += fabsf(ai0 - aS[j * 65 + ss]);
        rowS[ss * 64 + j] = acc;
    }
    // rows t = 1..63: D[s] = c + min(prev[s], prev[s-1], D[s-1])
    for (int t = 1; t < 64; ++t) {
        const float ait = aS[i * 65 + t];
        float diag = rowS[j];                      // prev[0]
        float left = fabsf(ait - aS[j * 65]) + diag;
        rowS[j] = left;
        for (int ss = 1; ss < 64; ++ss) {
            float up = rowS[ss * 64 + j];          // prev[s]
            float c  = fabsf(ait - aS[j * 65 + ss]);
            float D  = c + fminf(up, fminf(diag, left));
            rowS[ss * 64 + j] = D;
            diag = up; left = D;
        }
    }
    dtw_adj[i * 64 + j] = (i == j) ? 0.f : rowS[63 * 64 + j] * (1.f / 64.f);
}

// ---------------------------------------------------------------------------
// Kernel A: per-(b,s) fused attention + graph-conv diagonals.
// Never writes Q/K/V/att to HBM; outputs only d_att/d_dtw/d_avg [B][S*N].
// ---------------------------------------------------------------------------
__global__ __launch_bounds__(256) void attn_kernel(
    const float* __restrict__ x,
    const float* __restrict__ wq, const float* __restrict__ bq,
    const float* __restrict__ wk, const float* __restrict__ bk,
    const float* __restrict__ wv, const float* __restrict__ bv,
    const float* __restrict__ dtw_adj, const float* __restrict__ avg_adj,
    float* __restrict__ d_att, float* __restrict__ d_dtw, float* __restrict__ d_avg)
{
    __shared__ float lds[5 * 64 * LDA];           // ~83 KB of the 320 KB WGP LDS
    float* Xs = lds;
    float* Ws = lds + 1 * 64 * LDA;               // wq/wk/wv, later scores
    float* Qs = lds + 2 * 64 * LDA;               // Q, later att
    float* Ks = lds + 3 * 64 * LDA;
    float* Vs = lds + 4 * 64 * LDA;

    const int tid  = threadIdx.x;
    const int lane = tid & 31;
    const int wave = tid >> 5;
    const int bs   = blockIdx.x;                  // b*S + s
    const float* xb = x + (size_t)bs * 4096;

    for (int idx = tid; idx < 4096; idx += 256) { // stage x tile
        Xs[(idx >> 6) * LDA + (idx & 63)] = xb[idx];
    }

    const float* wmat[3]  = {wq, wk, wv};
    const float* bvec[3]  = {bq, bk, bv};
    float*       dst3[3]  = {Qs, Ks, Vs};
    for (int g = 0; g < 3; ++g) {                 // Q, K, V = x@W + b
        __syncthreads();
        for (int idx = tid; idx < 4096; idx += 256)
            Ws[(idx >> 6) * LDA + (idx & 63)] = wmat[g][idx];
        __syncthreads();
        for (int t = wave; t < 16; t += 8)
            wmma_tile_f32(Xs, LDA, Ws, LDA, false, dst3[g], LDA,
                          (t >> 2) * 16, (t & 3) * 16, lane, 1.f, bvec[g]);
    }
    __syncthreads();

    // scores = Q @ K^T / sqrt(64)  -> Ws
    for (int t = wave; t < 16; t += 8)
        wmma_tile_f32(Qs, LDA, Ks, LDA, true, Ws, LDA,
                      (t >> 2) * 16, (t & 3) * 16, lane, 0.125f, nullptr);
    __syncthreads();

    if (tid < 64) {                               // row softmax
        float mx = -1e30f;
        for (int c = 0; c < 64; ++c) mx = fmaxf(mx, Ws[tid * LDA + c]);
        float sum = 0.f;
        for (int c = 0; c < 64; ++c) {
            float e = expf(Ws[tid * LDA + c] - mx);
            Ws[tid * LDA + c] = e;
            sum += e;
        }
        float inv = 1.f / sum;
        for (int c = 0; c < 64; ++c) Ws[tid * LDA + c] *= inv;
    }
    __syncthreads();

    // att = P @ V -> Qs (Q dead)
    for (int t = wave; t < 16; t += 8)
        wmma_tile_f32(Ws, LDA, Vs, LDA, false, Qs, LDA,
                      (t >> 2) * 16, (t & 3) * 16, lane, 1.f, nullptr);
    __syncthreads();

    if (tid < 64) {                               // fused diag(x @ adj^T)
        const int n = tid;
        float da = 0.f, dd = 0.f, dv = 0.f;
        for (int f = 0; f < 64; ++f) {
            float xv = Xs[n * LDA + f];
            da += xv * Qs[n * LDA + f];
            dd += xv * dtw_adj[n * 64 + f];
            dv += xv * avg_adj[n * 64 + f];
        }
        d_att[bs * 64 + n] = da;
        d_dtw[bs * 64 + n] = dd;
        d_avg[bs * 64 + n] = dv;
    }
}

// ---------------------------------------------------------------------------
// Kernel C: stream W1 (201 MB) once; A_red[g][sn][h] = sum_j w_g[j]*W1[...,h].
// Bias projection written as per-sn partials (cb_part) -> deterministic
// fixed-order reduction later (no float atomics). Pure-bandwidth kernel;
// software prefetch keeps the load pipe primed.
// ---------------------------------------------------------------------------
__global__ __launch_bounds__(64) void reduce_w1_kernel(
    const float* __restrict__ W1,
    const float* __restrict__ w_att, const float* __restrict__ b_att,
    const float* __restrict__ w_dtw, const float* __restrict__ b_dtw,
    const float* __restrict__ w_avg, const float* __restrict__ b_avg,
    float* __restrict__ a_red, float* __restrict__ cb_part)
{
    __shared__ float wS[3][64];
    __shared__ float bS[3][64];
    const int h  = threadIdx.x;
    const int sn = blockIdx.x;
    wS[0][h] = w_att[h]; wS[1][h] = w_dtw[h]; wS[2][h] = w_avg[h];
    bS[0][h] = b_att[h]; bS[1][h] = b_dtw[h]; bS[2][h] = b_avg[h];
    __syncthreads();

    const float* Wb = W1 + (size_t)sn * 192 * 64;
    float accb = 0.f;
    for (int g = 0; g < 3; ++g) {
        float accw = 0.f;
        const float* Wg = Wb + g * 64 * 64;
        __builtin_prefetch(Wg + 64 * 64 + h, 0, 1);   // next 16KB chunk
        for (int jj = 0; jj < 64; ++jj) {
            float e = Wg[jj * 64 + h];                // coalesced 256B rows
            accw += wS[g][jj] * e;
            accb += bS[g][jj] * e;
        }
        a_red[((size_t)g * SN_ + sn) * 64 + h] = accw;
    }
    cb_part[sn * 64 + h] = accb;
}

// cbias[h] = sum_sn cb_part[sn][h], fixed order -> deterministic
__global__ __launch_bounds__(64) void cbias_kernel(
    const float* __restrict__ cb_part, float* __restrict__ cbias)
{
    const int h = threadIdx.x;
    float acc = 0.f;
    for (int sn = 0; sn < SN_; ++sn)
        acc += cb_part[sn * 64 + h];
    cbias[h] = acc;
}

// ---------------------------------------------------------------------------
// Kernel D: split-K WMMA contraction h_pre = sum_{g,sn} d_g[b,sn]*A_red[g,sn,h]
// Grid = 12 tiles x KPART partitions; each single-wave block runs a short
// 128-WMMA dependency chain and writes an independent partial (no atomics).
// ---------------------------------------------------------------------------
__global__ __launch_bounds__(32) void contract_kernel(
    const float* __restrict__ d_att, const float* __restrict__ d_dtw,
    const float* __restrict__ d_avg, const float* __restrict__ a_red,
    float* __restrict__ hpart)
{
    const int t    = blockIdx.x % 12;             // tile: 3 M-tiles x 4 N-tiles
    const int p    = blockIdx.x / 12;             // K partition 0..KPART-1
    const int g    = p >> 3;                      // adjacency kind
    const int kr   = p & 7;                       // 1/8 of the 4096 K range
    const int m0   = (t >> 2) * 16;
    const int n0   = (t & 3) * 16;
    const int lane = threadIdx.x;
    const int half = (lane < 16) ? 0 : 2;
    const int l15  = lane & 15;
    const int m = m0 + l15, n = n0 + l15;

    const float* dg = (g == 0) ? d_att : (g == 1) ? d_dtw : d_avg;
    const float* Bg = a_red + (size_t)g * SN_ * 64;

    v8f acc = {};
    const int kbeg = kr * 512, kend = kbeg + 512;
    for (int k0 = kbeg; k0 < kend; k0 += 4) {
        v2f a, b;
        a.x = dg[m * SN_ + k0 + half];
        a.y = dg[m * SN_ + k0 + half + 1];
        b.x = Bg[(k0 + half) * 64 + n];
        b.y = Bg[(k0 + half + 1) * 64 + n];
        acc = __builtin_amdgcn_wmma_f32_16x16x4_f32(
                  false, a, false, b, (short)0, acc, false, false);
    }
#pragma unroll
    for (int r = 0; r < 8; ++r) {
        int mm = m0 + r + ((lane < 16) ? 0 : 8);   // mm <= 47 == B_-1
        hpart[(size_t)p * (B_ * 64) + mm * 64 + n] = acc[r];
    }
}

// h[b,h] = relu( sum_p hpart[p][b][h] + cbias[h] + b1[h] ), fixed order
__global__ __launch_bounds__(256) void reduce_h_kernel(
    const float* __restrict__ hpart, const float* __restrict__ cbias,
    const float* __restrict__ b1, float* __restrict__ hbuf)
{
    const int idx = blockIdx.x * 256 + threadIdx.x;   // 48*64
    if (idx >= B_ * 64) return;
    const int n = idx & 63;
    float acc = cbias[n] + b1[n];
    for (int p = 0; p < KPART; ++p)
        acc += hpart[(size_t)p * (B_ * 64) + idx];
    hbuf[idx] = fmaxf(acc, 0.f);
}

// ---------------------------------------------------------------------------
// Kernel E: out[b, c] = h[b,:] @ W2[:,c] + b2[c]   (48 x 768, tiny)
// ---------------------------------------------------------------------------
__global__ __launch_bounds__(256) void out_kernel(
    const float* __restrict__ hbuf, const float* __restrict__ W2,
    const float* __restrict__ b2, float* __restrict__ out)
{
    const int idx = blockIdx.x * 256 + threadIdx.x;
    if (idx >= B_ * N_ * HOR_) return;
    const int b = idx / (N_ * HOR_), c = idx % (N_ * HOR_);
    float acc = b2[c];
    for (int h = 0; h < 64; ++h)
        acc += hbuf[b * 64 + h] * W2[h * (N_ * HOR_) + c];
    out[idx] = acc;
}

extern "C" void kernel_launch(void* const* d_in, const int* in_sizes, int n_in,
                              void* d_out, int out_size, void* d_ws, size_t ws_size,
                              hipStream_t stream)
{
    (void)in_sizes; (void)n_in; (void)out_size; (void)ws_size;
    const float* x     = (const float*)d_in[0];
    const float* wq    = (const float*)d_in[1];
    const float* bq    = (const float*)d_in[2];
    const float* wk    = (const float*)d_in[3];
    const float* bk    = (const float*)d_in[4];
    const float* wv    = (const float*)d_in[5];
    const float* bv    = (const float*)d_in[6];
    const float* w_att = (const float*)d_in[7];
    const float* b_att = (const float*)d_in[8];
    const float* w_dtw = (const float*)d_in[9];
    const float* b_dtw = (const float*)d_in[10];
    const float* w_avg = (const float*)d_in[11];
    const float* b_avg = (const float*)d_in[12];
    const float* W1    = (const float*)d_in[13];
    const float* b1    = (const float*)d_in[14];
    const float* W2    = (const float*)d_in[15];
    const float* b2    = (const float*)d_in[16];
    float* out = (float*)d_out;

    float* ws      = (float*)d_ws;
    float* dtw_adj = ws;                       // 4096
    float* avg_adj = dtw_adj + 4096;           // 4096
    float* d_att   = avg_adj + 4096;           // 48*4096
    float* d_dtw   = d_att + B_ * SN_;
    float* d_avg   = d_dtw + B_ * SN_;
    float* a_red   = d_avg + B_ * SN_;         // 3*4096*64
    float* cb_part = a_red + 3 * SN_ * 64;     // 4096*64
    float* cbias   = cb_part + SN_ * 64;       // 64
    float* hpart   = cbias + 64;               // KPART*48*64
    float* hbuf    = hpart + KPART * B_ * 64;  // 48*64

    dtw_kernel<<<64, 64, 0, stream>>>(x, dtw_adj, avg_adj);
    attn_kernel<<<B_ * S_, 256, 0, stream>>>(x, wq, bq, wk, bk, wv, bv,
                                             dtw_adj, avg_adj,
                                             d_att, d_dtw, d_avg);
    reduce_w1_kernel<<<SN_, 64, 0, stream>>>(W1, w_att, b_att, w_dtw, b_dtw,
                                             w_avg, b_avg, a_red, cb_part);
    cbias_kernel<<<1, 64, 0, stream>>>(cb_part, cbias);
    contract_kernel<<<12 * KPART, 32, 0, stream>>>(d_att, d_dtw, d_avg,
                                                   a_red, hpart);
    reduce_h_kernel<<<(B_ * 64 + 255) / 256, 256, 0, stream>>>(hpart, cbias,
                                                               b1, hbuf);
    out_kernel<<<(B_ * N_ * HOR_ + 255) / 256, 256, 0, stream>>>(hbuf, W2, b2, out);
}